// Texture_Renderer_74259984548529
// MI455X (gfx1250) — compile-verified
//
#include <hip/hip_runtime.h>

typedef __attribute__((ext_vector_type(2))) float v2f;
typedef __attribute__((ext_vector_type(8))) float v8f;

#define BATCH 4
#define HGT   512
#define WID   512
#define HW    (HGT * WID)          // 262144
#define TEXW  2048
#define TEXHW (TEXW * TEXW)

// SH constants (match reference SH_CONST)
__device__ const float SHC[9] = {
    0.28209479177387814f,   // 1/sqrt(4pi)
    1.0233267079464885f,    // 2pi/3 * sqrt(3/(4pi))
    1.0233267079464885f,
    1.0233267079464885f,
    0.8580855308097834f,    // pi/4 * 3 * sqrt(5/(12pi))
    0.8580855308097834f,
    0.8580855308097834f,
    0.4290427654048917f,    // pi/4 * 3/2 * sqrt(5/(12pi))
    0.24770795610037571f    // pi/4 * 1/2 * sqrt(5/(4pi))
};

__global__ __launch_bounds__(256)
void Texture_Renderer_kernel(const float* __restrict__ fa,    // (B,F,3,9)
                             const float* __restrict__ bary,  // (B,H,W,1,3)
                             const int*   __restrict__ p2f,   // (B,H,W,1)
                             const float* __restrict__ tex,   // (1,3,2048,2048)
                             const float* __restrict__ lp,    // (B,9,3)
                             float*       __restrict__ out)   // 3 x (B,3,H,W)
{
    // Per-batch SH coefficient matrix, folded with SHC, padded to 16x12 for WMMA A.
    __shared__ float sWc[16][12];

    const int b        = blockIdx.x >> 10;                     // 1024 blocks per image
    const int pixInImg = ((blockIdx.x & 1023) << 8) + threadIdx.x;
    const int p        = b * HW + pixInImg;

    {
        int t = threadIdx.x;
        if (t < 16 * 12) {
            int m = t / 12, n = t % 12;
            float w = 0.0f;
            if (m < 3 && n < 9) w = lp[b * 27 + n * 3 + m] * SHC[n];
            sWc[m][n] = w;
        }
    }
    __syncthreads();

    // ---- gather + barycentric interpolation -------------------------------
    const int  pf  = p2f[p];
    const bool bg  = (pf < 0);
    const int  idx = bg ? 0 : pf;
    const float* __restrict__ arow = fa + (size_t)idx * 27;    // attrs[(B*F),3,9] row
    __builtin_prefetch(arow, 0, 1);                            // global_prefetch_b8
    __builtin_prefetch(arow + 16, 0, 1);

    const float b0 = bary[(size_t)p * 3 + 0];
    const float b1 = bary[(size_t)p * 3 + 1];
    const float b2 = bary[(size_t)p * 3 + 2];

    float attr[9];
#pragma unroll
    for (int d = 0; d < 9; ++d) {
        float v = b0 * arow[d] + b1 * arow[9 + d] + b2 * arow[18 + d];
        attr[d] = bg ? 0.0f : v;
    }
    const float vis = bg ? 0.0f : 1.0f;

    // ---- SH basis (12-padded for K chunks of 4) ---------------------------
    const float nx = attr[3], ny = attr[4], nz = attr[5];
    float shv[12];
    shv[0] = 1.0f;       shv[1] = nx;          shv[2]  = ny;   shv[3]  = nz;
    shv[4] = nx * ny;    shv[5] = nx * nz;     shv[6]  = ny * nz;
    shv[7] = nx * nx - ny * ny;
    shv[8] = 3.0f * nz * nz - 1.0f;
    shv[9] = 0.0f;       shv[10] = 0.0f;       shv[11] = 0.0f;

    // ---- SH einsum on the matrix pipe: shading(3xN) = Wc(3x12) * basis(12xN)
    // Per wave: two 16-pixel groups, 3 chunks of K=4 -> 6x v_wmma_f32_16x16x4_f32
    const int lane = threadIdx.x & 31;
    const int half = lane >> 4;     // 0: K=0,1 slots | 1: K=2,3 slots
    const int lrow = lane & 15;

    v8f acc0 = {};
    v8f acc1 = {};
#pragma unroll
    for (int k = 0; k < 3; ++k) {
        // A 16x4 f32 layout: lane m holds M=m; VGPR0/1 = K0,K1 (lanes 0-15), K2,K3 (lanes 16-31)
        v2f a;
        a.x = sWc[lrow][4 * k + 2 * half + 0];
        a.y = sWc[lrow][4 * k + 2 * half + 1];
#pragma unroll
        for (int g = 0; g < 2; ++g) {
            const int src = (g << 4) | lrow;   // pixel owning column N=lrow of group g
            float x0 = __shfl(shv[4 * k + 0], src, 32);
            float x1 = __shfl(shv[4 * k + 1], src, 32);
            float x2 = __shfl(shv[4 * k + 2], src, 32);
            float x3 = __shfl(shv[4 * k + 3], src, 32);
            // B 4x16 f32 layout: VGPR0 = K0 (lanes 0-15) / K2 (lanes 16-31), VGPR1 = K1/K3
            v2f bb;
            bb.x = half ? x2 : x0;
            bb.y = half ? x3 : x1;
            if (g == 0)
                acc0 = __builtin_amdgcn_wmma_f32_16x16x4_f32(
                    false, a, false, bb, (short)0, acc0, false, false);
            else
                acc1 = __builtin_amdgcn_wmma_f32_16x16x4_f32(
                    false, a, false, bb, (short)0, acc1, false, false);
        }
    }
    // Read back D: row M=c lives in VGPR c, column N at lane N (lanes 0-15)
    float shade[3];
#pragma unroll
    for (int c = 0; c < 3; ++c) {
        float s0 = __shfl(acc0[c], lrow, 32);
        float s1 = __shfl(acc1[c], lrow, 32);
        shade[c] = half ? s1 : s0;
    }

    // ---- bilinear texture sample (zero-padded OOB, align_corners=False) ---
    const float u = attr[0], vv = attr[1];
    const float x = (u  + 1.0f) * (0.5f * (float)TEXW) - 0.5f;
    const float y = (vv + 1.0f) * (0.5f * (float)TEXW) - 0.5f;
    const float x0f = floorf(x), y0f = floorf(y);
    const int   x0i = (int)x0f,  y0i = (int)y0f;
    const int   x1i = x0i + 1,   y1i = y0i + 1;
    const float wx1 = x - x0f, wx0 = 1.0f - wx1;
    const float wy1 = y - y0f, wy0 = 1.0f - wy1;

    const float m00 = (x0i >= 0 && x0i < TEXW && y0i >= 0 && y0i < TEXW) ? 1.0f : 0.0f;
    const float m10 = (x1i >= 0 && x1i < TEXW && y0i >= 0 && y0i < TEXW) ? 1.0f : 0.0f;
    const float m01 = (x0i >= 0 && x0i < TEXW && y1i >= 0 && y1i < TEXW) ? 1.0f : 0.0f;
    const float m11 = (x1i >= 0 && x1i < TEXW && y1i >= 0 && y1i < TEXW) ? 1.0f : 0.0f;

    const int cx0 = min(max(x0i, 0), TEXW - 1);
    const int cx1 = min(max(x1i, 0), TEXW - 1);
    const int cy0 = min(max(y0i, 0), TEXW - 1);
    const int cy1 = min(max(y1i, 0), TEXW - 1);

    const float w00 = wx0 * wy0, w10 = wx1 * wy0, w01 = wx0 * wy1, w11 = wx1 * wy1;

    float alb[3];
#pragma unroll
    for (int c = 0; c < 3; ++c) {
        const float* __restrict__ tc = tex + (size_t)c * TEXHW;
        float t00 = tc[(size_t)cy0 * TEXW + cx0] * m00;
        float t10 = tc[(size_t)cy0 * TEXW + cx1] * m10;
        float t01 = tc[(size_t)cy1 * TEXW + cx0] * m01;
        float t11 = tc[(size_t)cy1 * TEXW + cx1] * m11;
        alb[c] = t00 * w00 + t10 * w10 + t01 * w01 + t11 * w11;
    }

    // ---- outputs (coalesced) ---------------------------------------------
    const size_t o1    = (size_t)(b * 3) * HW + pixInImg;
    const size_t osz   = (size_t)BATCH * 3 * HW;       // one output tensor
    const float  maskA = (vis > 0.0f) ? 1.0f : 0.0f;
#pragma unroll
    for (int c = 0; c < 3; ++c) {
        out[o1 + (size_t)c * HW]             = alb[c] * alb[c] * shade[c]; // images*albedo
        out[osz + o1 + (size_t)c * HW]       = maskA;                      // mask_all
        out[2 * osz + o1 + (size_t)c * HW]   = (attr[6 + c] > 0.0f) ? 1.0f : 0.0f;
    }
}

extern "C" void kernel_launch(void* const* d_in, const int* in_sizes, int n_in,
                              void* d_out, int out_size, void* d_ws, size_t ws_size,
                              hipStream_t stream) {
    (void)in_sizes; (void)n_in; (void)out_size; (void)d_ws; (void)ws_size;
    const float* fa   = (const float*)d_in[0];   // face_attributes
    const float* bary = (const float*)d_in[1];   // bary_coords
    const int*   p2f  = (const int*)  d_in[2];   // pix_to_face
    // d_in[3] = zbuf (unused by reference output)
    const float* tex  = (const float*)d_in[4];   // texture_images
    const float* lp   = (const float*)d_in[5];   // light_params
    float* out = (float*)d_out;

    dim3 grid(BATCH * (HW / 256));   // 4096 blocks
    dim3 block(256);
    Texture_Renderer_kernel<<<grid, block, 0, stream>>>(fa, bary, p2f, tex, lp, out);
}